// GCN_77421080478455
// MI455X (gfx1250) — compile-verified
//
#include <hip/hip_runtime.h>
#include <hip/hip_bf16.h>

typedef __attribute__((ext_vector_type(2))) float v2f;
typedef __attribute__((ext_vector_type(8))) float v8f;

// ---------------------------------------------------------------------------
// init: deg = 1 (self loop), zero both aggregation buffers
// ---------------------------------------------------------------------------
__global__ void k_init(float* __restrict__ deg, float* __restrict__ aggB,
                       float* __restrict__ aggC, int n) {
  int idx = blockIdx.x * blockDim.x + threadIdx.x;
  if (idx < n) deg[idx] = 1.0f;
  if (idx < n * 16) { aggB[idx] = 0.0f; aggC[idx] = 0.0f; }
}

// ---------------------------------------------------------------------------
// degree accumulation: deg[dst[e]] += 1
// ---------------------------------------------------------------------------
__global__ void k_deg(const int* __restrict__ dst, float* __restrict__ deg, int E) {
  int e = blockIdx.x * blockDim.x + threadIdx.x;
  if (e < E) atomicAdd(&deg[dst[e]], 1.0f);
}

// ---------------------------------------------------------------------------
// dinv = rsqrt(deg)
// ---------------------------------------------------------------------------
__global__ void k_rsqrt(const float* __restrict__ deg, float* __restrict__ dinv, int n) {
  int i = blockIdx.x * blockDim.x + threadIdx.x;
  if (i < n) dinv[i] = rsqrtf(deg[i]);
}

// ---------------------------------------------------------------------------
// WMMA GEMM: H[N x 16] = A[N x K] @ W[K x 16], one wave32 per 16-row tile.
// Uses V_WMMA_F32_16X16X4_F32, K/4 steps, W staged in LDS.
//
// A layout (16x4 f32): lanes 0-15 hold rows M=0..15 with K=k0+0 (v0) / k0+1 (v1);
//                      lanes 16-31 hold K=k0+2 / k0+3.
// B layout (4x16 f32): VGPR j: lanes 0-15 = row K=k0+j, lanes 16-31 = row K=k0+j+2.
// C/D layout (16x16 f32): VGPR i: lanes 0-15 -> M=i, lanes 16-31 -> M=i+8, N=lane&15.
// ---------------------------------------------------------------------------
template <int K>
__global__ void k_gemm_wmma(const float* __restrict__ A, const float* __restrict__ W,
                            float* __restrict__ H, int ntiles) {
  __shared__ float Ws[K * 16];
  for (int t = threadIdx.x; t < K * 16; t += blockDim.x) Ws[t] = W[t];
  __syncthreads();

  const int lane = threadIdx.x & 31;
  const int tile = blockIdx.x * (blockDim.x >> 5) + (threadIdx.x >> 5);
  if (tile >= ntiles) return;  // wave-uniform: EXEC stays all-ones for WMMA

  const int half = lane >> 4;   // 0 for lanes 0-15, 1 for lanes 16-31
  const int mn   = lane & 15;   // row (for A) / column (for B,C,D)
  const int koff = half * 2;

  const float* arow = A + (size_t)(tile * 16 + mn) * K;

  v8f c = {0.f, 0.f, 0.f, 0.f, 0.f, 0.f, 0.f, 0.f};
#pragma unroll
  for (int k0 = 0; k0 < K; k0 += 4) {
    v2f a = *(const v2f*)(arow + k0 + koff);          // global_load_b64
    v2f b;
    b.x = Ws[(k0 + koff + 0) * 16 + mn];              // ds_load_b32
    b.y = Ws[(k0 + koff + 1) * 16 + mn];
    c = __builtin_amdgcn_wmma_f32_16x16x4_f32(
        /*neg_a=*/false, a, /*neg_b=*/false, b,
        /*c_mod=*/(short)0, c, /*reuse_a=*/false, /*reuse_b=*/false);
  }

  // scatter 8 accumulator rows back: row = tile*16 + half*8 + i, col = mn
  float* hrow = H + (size_t)tile * 256 + half * 128 + mn;
#pragma unroll
  for (int i = 0; i < 8; i++) hrow[i * 16] = c[i];
}

// ---------------------------------------------------------------------------
// edge scatter: AGG[dst] += dinv[src]*dinv[dst] * H[src], 16 lanes per edge
// ---------------------------------------------------------------------------
__global__ void k_scatter(const int* __restrict__ src, const int* __restrict__ dst,
                          const float* __restrict__ dinv, const float* __restrict__ H,
                          float* __restrict__ AGG, int E) {
  int idx = blockIdx.x * blockDim.x + threadIdx.x;
  int e = idx >> 4;
  int j = idx & 15;
  if (e >= E) return;
  int s = src[e];
  int d = dst[e];
  float norm = dinv[s] * dinv[d];
  atomicAdd(&AGG[d * 16 + j], norm * H[s * 16 + j]);
}

// ---------------------------------------------------------------------------
// combine: h = relu(AGG + dinv^2 * H + bias)   (self-loop term + bias + ReLU)
// ---------------------------------------------------------------------------
__global__ void k_combine(const float* __restrict__ AGG, const float* __restrict__ H,
                          const float* __restrict__ dinv, const float* __restrict__ bias,
                          float* __restrict__ out, int n) {
  int idx = blockIdx.x * blockDim.x + threadIdx.x;
  int i = idx >> 4;
  int j = idx & 15;
  if (i >= n) return;
  float di = dinv[i];
  out[idx] = fmaxf(AGG[idx] + di * di * H[idx] + bias[j], 0.0f);
}

// ---------------------------------------------------------------------------
// final: out[i] = sum_j relu(G[i,j] + l1b[j]) * l2W[j] + l2b
// 16 lanes per node, shuffle reduction within the 16-lane group
// ---------------------------------------------------------------------------
__global__ void k_final(const float* __restrict__ G, const float* __restrict__ l1b,
                        const float* __restrict__ l2W, const float* __restrict__ l2b,
                        float* __restrict__ out, int n) {
  int idx = blockIdx.x * blockDim.x + threadIdx.x;
  int i = idx >> 4;
  int j = idx & 15;
  float v = 0.0f;
  if (i < n) v = fmaxf(G[idx] + l1b[j], 0.0f) * l2W[j];
#pragma unroll
  for (int off = 8; off; off >>= 1) v += __shfl_xor(v, off, 16);
  if (i < n && j == 0) out[i] = v + l2b[0];
}

// ---------------------------------------------------------------------------
extern "C" void kernel_launch(void* const* d_in, const int* in_sizes, int n_in,
                              void* d_out, int out_size, void* d_ws, size_t ws_size,
                              hipStream_t stream) {
  const float* x   = (const float*)d_in[0];
  const int*   ei  = (const int*)d_in[1];   // [2, E] int32 (jax x64 disabled)
  const float* W1  = (const float*)d_in[2];
  const float* b1  = (const float*)d_in[3];
  const float* W2  = (const float*)d_in[4];
  const float* b2  = (const float*)d_in[5];
  const float* l1W = (const float*)d_in[6];
  const float* l1b = (const float*)d_in[7];
  const float* l2W = (const float*)d_in[8];
  const float* l2b = (const float*)d_in[9];

  const int N = in_sizes[0] / 128;   // 100000 (divisible by 16)
  const int E = in_sizes[1] / 2;     // 3200000
  const int* src = ei;
  const int* dst = ei + E;

  // workspace layout: deg[N] dinv[N] bufA[N*16] bufB[N*16] bufC[N*16]  (~20 MB)
  float* deg  = (float*)d_ws;
  float* dinv = deg + N;
  float* bufA = dinv + N;                 // GEMM output H
  float* bufB = bufA + (size_t)N * 16;    // AGG1 -> h1
  float* bufC = bufB + (size_t)N * 16;    // AGG2 -> h2

  const int ntiles = N / 16;                             // 6250 waves per GEMM
  const int gemmBlocks = (ntiles * 32 + 255) / 256;      // 8 waves / block
  const int nodeElems  = N * 16;
  const int nodeBlocks = (nodeElems + 255) / 256;
  const int edgeBlocks = (E + 255) / 256;
  const int scatBlocks = (int)(((size_t)E * 16 + 255) / 256);

  // degrees + normalization
  k_init  <<<nodeBlocks, 256, 0, stream>>>(deg, bufB, bufC, N);
  k_deg   <<<edgeBlocks, 256, 0, stream>>>(dst, deg, E);
  k_rsqrt <<<(N + 255) / 256, 256, 0, stream>>>(deg, dinv, N);

  // layer 1: H = x @ W1 ; AGG over edges ; h1 = relu(AGG + dinv^2 H + b1)
  k_gemm_wmma<128><<<gemmBlocks, 256, 0, stream>>>(x, W1, bufA, ntiles);
  k_scatter<<<scatBlocks, 256, 0, stream>>>(src, dst, dinv, bufA, bufB, E);
  k_combine<<<nodeBlocks, 256, 0, stream>>>(bufB, bufA, dinv, b1, bufB, N);

  // layer 2: H = h1 @ W2 ; AGG ; h2 = relu(...)
  k_gemm_wmma<16><<<gemmBlocks, 256, 0, stream>>>(bufB, W2, bufA, ntiles);
  k_scatter<<<scatBlocks, 256, 0, stream>>>(src, dst, dinv, bufA, bufC, E);
  k_combine<<<nodeBlocks, 256, 0, stream>>>(bufC, bufA, dinv, b2, bufC, N);

  // head: G = h2 @ lin1_W ; out = relu(G + l1b) @ lin2_W + l2b
  k_gemm_wmma<16><<<gemmBlocks, 256, 0, stream>>>(bufC, l1W, bufA, ntiles);
  k_final<<<nodeBlocks, 256, 0, stream>>>(bufA, l1b, l2W, l2b, (float*)d_out, N);
}